// DeepRNN_13623636263695
// MI455X (gfx1250) — compile-verified
//
#include <hip/hip_runtime.h>
#include <hip/hip_bf16.h>

typedef __attribute__((ext_vector_type(2))) float v2f;
typedef __attribute__((ext_vector_type(8))) float v8f;

#define BDIM 32
#define SDIM 2048
#define HDIM 512
#define LDIM 4
#define MROWS (BDIM * SDIM)   // 65536 rows in the bs-flattened GEMMs
#define NBLK_SCAN 16          // persistent blocks in the recurrent scan

// ---------------------------------------------------------------------------
// Kernel 1: pre[row, n] = sum_k x[row,k] * W[n,k] + (b_ih[n] + b_hh[n])
// row = b*S + s  (65536 rows), n in [0,512), K = 256 (layer 0) or 512.
// Block = 128 threads (4 waves). Block tile 64(M) x 64(N); each wave owns a
// 16(M) x 64(N) stripe = 4 accumulator tiles sharing one A fragment.
// f32 WMMA fragment layout (16x16x4): lanes 0-15 hold K=k,k+1; lanes 16-31
// hold K=k+2,k+3 -> a single 8B float2 load per lane per K-step.
// ---------------------------------------------------------------------------
__global__ __launch_bounds__(128) void pre_gemm_kernel(
    const float* __restrict__ x, const float* __restrict__ W,
    const float* __restrict__ bih, const float* __restrict__ bhh,
    float* __restrict__ pre, int K)
{
    const int lane = threadIdx.x & 31;
    const int wave = threadIdx.x >> 5;
    const int half = lane >> 4;   // 0: K=k,k+1   1: K=k+2,k+3
    const int l16  = lane & 15;

    const int row0 = (blockIdx.x << 6) + (wave << 4);  // 16-row stripe of this wave
    const int col0 = (blockIdx.y << 6);                // 64-col stripe of this block

    const float* xrow = x + (size_t)(row0 + l16) * (size_t)K;

    v8f acc[4] = {};

    for (int k = 0; k < K; k += 4) {
        const int ka = k + 2 * half;
        v2f a = *reinterpret_cast<const v2f*>(xrow + ka);
#pragma unroll
        for (int ct = 0; ct < 4; ++ct) {
            const int n = col0 + ct * 16 + l16;
            v2f b = *reinterpret_cast<const v2f*>(W + (size_t)n * (size_t)K + ka);
            acc[ct] = __builtin_amdgcn_wmma_f32_16x16x4_f32(
                false, a, false, b, (short)0, acc[ct], false, false);
        }
    }

#pragma unroll
    for (int ct = 0; ct < 4; ++ct) {
        const int n = col0 + ct * 16 + l16;
        const float bias = bih[n] + bhh[n];
#pragma unroll
        for (int j = 0; j < 8; ++j) {
            const int m = half ? (j + 8) : j;  // C layout: lanes16-31 hold M=j+8
            pre[(size_t)(row0 + m) * HDIM + n] = acc[ct][j] + bias;
        }
    }
}

// ---------------------------------------------------------------------------
// Per-layer init: zero grid-barrier counter, copy h0[layer] into h ping buffer.
// ---------------------------------------------------------------------------
__global__ void init_layer_kernel(float* __restrict__ hbuf0,
                                  const float* __restrict__ h0l,
                                  unsigned* __restrict__ bar)
{
    const int i = blockIdx.x * blockDim.x + threadIdx.x;
    if (i == 0) *bar = 0u;
    if (i < BDIM * HDIM) hbuf0[i] = h0l[i];
}

// ---------------------------------------------------------------------------
// Kernel 2: recurrent scan  h_t = relu(h_{t-1} @ Whh^T + pre_t), in place:
// preys[b,t,:] is read as pre_t and overwritten with ys_t.
// 16 persistent blocks x 128 threads; block g owns H-columns [32g, 32g+32).
// Whh slice staged transposed in LDS (stride 33 floats -> no bank conflicts).
// Two interleaved WMMA accumulators break the 128-deep serial K chain.
// Cross-block sync: release/acquire atomic grid barrier (16 resident blocks).
// ---------------------------------------------------------------------------
__global__ __launch_bounds__(128) void rnn_scan_kernel(
    float* __restrict__ preys,        // [B,S,H] in: pre, out: ys
    const float* __restrict__ Whh,    // [H,H] this layer, row-major [n][k]
    float* __restrict__ hbuf0,        // [B,H] ping (holds h0 on entry)
    float* __restrict__ hbuf1,        // [B,H] pong
    float* __restrict__ hfinal,       // [B,H] final hidden out
    unsigned* __restrict__ bar)
{
    __shared__ float WslT[HDIM][33];  // [k][n_local], padded stride

    const int g    = blockIdx.x;
    const int tid  = threadIdx.x;
    const int lane = tid & 31;
    const int wave = tid >> 5;
    const int half = lane >> 4;
    const int l16  = lane & 15;
    const int r    = wave & 1;        // row tile (B rows 0-15 / 16-31)
    const int c    = wave >> 1;       // col tile within the 32-col slice
    const int nloc = c * 16 + l16;
    const int ngl  = g * 32 + nloc;   // global H column
    const int arow = r * 16 + l16;    // A-fragment row (b index)

    // Stage Whh^T slice: WslT[k][n] = Whh[32g+n][k]; coalesced global reads.
    for (int idx = tid; idx < 32 * HDIM; idx += 128) {
        const int nl = idx >> 9;      // / 512
        const int k  = idx & 511;
        WslT[k][nl] = Whh[(size_t)(g * 32 + nl) * HDIM + k];
    }
    __syncthreads();

    for (int t = 0; t < SDIM; ++t) {
        const float* hin  = (t & 1) ? hbuf1 : hbuf0;
        float*       hout = (t & 1) ? hbuf0 : hbuf1;

        v8f acc0 = {}, acc1 = {};
        const float* ha = hin + (size_t)arow * HDIM;

#pragma unroll 4
        for (int k = 0; k < HDIM; k += 8) {
            const int ka0 = k + 2 * half;
            const int ka1 = k + 4 + 2 * half;
            v2f a0 = *reinterpret_cast<const v2f*>(ha + ka0);
            v2f a1 = *reinterpret_cast<const v2f*>(ha + ka1);
            v2f b0; b0.x = WslT[ka0][nloc]; b0.y = WslT[ka0 + 1][nloc];
            v2f b1; b1.x = WslT[ka1][nloc]; b1.y = WslT[ka1 + 1][nloc];
            acc0 = __builtin_amdgcn_wmma_f32_16x16x4_f32(
                false, a0, false, b0, (short)0, acc0, false, false);
            acc1 = __builtin_amdgcn_wmma_f32_16x16x4_f32(
                false, a1, false, b1, (short)0, acc1, false, false);
        }

#pragma unroll
        for (int j = 0; j < 8; ++j) {
            const int m     = half ? (j + 8) : j;
            const int b_row = r * 16 + m;
            const size_t idx = ((size_t)b_row * SDIM + (size_t)t) * HDIM + ngl;
            float v = acc0[j] + acc1[j] + preys[idx];
            v = fmaxf(v, 0.0f);
            preys[idx] = v;                              // ys_t (in place over pre)
            hout[(size_t)b_row * HDIM + ngl] = v;        // h_t for next step
            if (t == SDIM - 1) hfinal[(size_t)b_row * HDIM + ngl] = v;
        }

        // ---- grid barrier: all 16 blocks finish step t before step t+1 ----
        __threadfence();                   // release h_t writes (device scope)
        __syncthreads();
        if (tid == 0) {
            __hip_atomic_fetch_add(bar, 1u, __ATOMIC_ACQ_REL,
                                   __HIP_MEMORY_SCOPE_AGENT);
            const unsigned target = (unsigned)NBLK_SCAN * (unsigned)(t + 1);
            while (__hip_atomic_load(bar, __ATOMIC_ACQUIRE,
                                     __HIP_MEMORY_SCOPE_AGENT) < target) {
                __builtin_amdgcn_s_sleep(2);
            }
        }
        __syncthreads();
        __threadfence();                   // acquire: invalidate stale h lines
    }
}

// ---------------------------------------------------------------------------
// Host-side orchestration (all launches on `stream`; graph-capture safe).
// Workspace: [bar:256B][buf0:BSH f32][buf1:BSH f32][hb0][hb1]  (~256 MB)
// Layer chaining: L0: x=in  -> buf0 ; L1: buf0 -> buf1 ;
//                 L2: buf1 -> buf0 ; L3: buf0 -> d_out (ys region).
// pre is written into the same buffer that the scan converts in place to ys.
// ---------------------------------------------------------------------------
extern "C" void kernel_launch(void* const* d_in, const int* in_sizes, int n_in,
                              void* d_out, int out_size, void* d_ws, size_t ws_size,
                              hipStream_t stream)
{
    (void)in_sizes; (void)n_in; (void)out_size; (void)ws_size;

    const float* x_in = (const float*)d_in[0];  // [B,S,I]
    const float* h0   = (const float*)d_in[1];  // [L,B,H]
    const float* wih0 = (const float*)d_in[2];  // [H,I]
    const float* wihn = (const float*)d_in[3];  // [L-1,H,H]
    const float* whh  = (const float*)d_in[4];  // [L,H,H]
    const float* bih  = (const float*)d_in[5];  // [L,H]
    const float* bhh  = (const float*)d_in[6];  // [L,H]
    float* out = (float*)d_out;                 // [B,S,H] ++ [L,B,H]

    const size_t BSH = (size_t)BDIM * SDIM * HDIM;   // 33,554,432 floats
    const size_t BH  = (size_t)BDIM * HDIM;          // 16,384 floats

    unsigned* bar = (unsigned*)d_ws;
    float* buf0 = (float*)((char*)d_ws + 256);
    float* buf1 = buf0 + BSH;
    float* hb0  = buf1 + BSH;
    float* hb1  = hb0 + BH;
    float* hfin = out + BSH;                         // final hidden states region

    const dim3 blk(128);
    const dim3 grid_gemm(MROWS / 64, HDIM / 64);     // 1024 x 8 blocks

    for (int layer = 0; layer < LDIM; ++layer) {
        const float* x; int K; float* pre;
        switch (layer) {
            case 0:  x = x_in; K = 256; pre = buf0; break;
            case 1:  x = buf0; K = 512; pre = buf1; break;
            case 2:  x = buf1; K = 512; pre = buf0; break;
            default: x = buf0; K = 512; pre = out;  break;  // layer 3 -> d_out
        }
        const float* W = (layer == 0)
            ? wih0 : (wihn + (size_t)(layer - 1) * HDIM * HDIM);

        pre_gemm_kernel<<<grid_gemm, blk, 0, stream>>>(
            x, W, bih + (size_t)layer * HDIM, bhh + (size_t)layer * HDIM, pre, K);

        init_layer_kernel<<<(BDIM * HDIM + 255) / 256, 256, 0, stream>>>(
            hb0, h0 + (size_t)layer * BH, bar);

        rnn_scan_kernel<<<NBLK_SCAN, blk, 0, stream>>>(
            pre, whh + (size_t)layer * HDIM * HDIM, hb0, hb1,
            hfin + (size_t)layer * BH, bar);
    }
}